// EvalEig_27994596835952
// MI455X (gfx1250) — compile-verified
//
#include <hip/hip_runtime.h>
#include <stdint.h>

// Problem constants (match reference: RN=10000 grid points, NE=1000 eigenvalues)
#define RN     10000
#define NE     1000
#define CHUNK  100                      // intervals per chunk
#define NCHUNK 100                      // ceil((RN-1)/CHUNK)
#define NPAIR  (NCHUNK * NE)            // parallel (chunk, eigenvalue) work items
#define RROW   (CHUNK + 4)              // padded LDS row of r values

// ---------------------------------------------------------------------------
// RK4 single column of y'' = c(t) * y, with the three c evaluations shared
// between columns (c1 = c(t), c2 = c(t+h/2), c3 = c(t+h)).
// ---------------------------------------------------------------------------
__device__ __forceinline__ void rk4_col(float c1, float c2, float c3, float h,
                                        float& y, float& p) {
  float k1y = p;
  float k1p = c1 * y;
  float k2y = p + 0.5f * h * k1p;
  float k2p = c2 * (y + 0.5f * h * k1y);
  float k3y = p + 0.5f * h * k2p;
  float k3p = c2 * (y + 0.5f * h * k2y);
  float k4y = p + h * k3p;
  float k4p = c3 * (y + h * k3y);
  float s = h * (1.0f / 6.0f);
  y += s * (k1y + 2.0f * (k2y + k3y) + k4y);
  p += s * (k1p + 2.0f * (k2p + k3p) + k4p);
}

// Advance NCOL columns across one grid interval [ta, ta+dr] with substepping.
// Substep count depends only on ta -> identical in phase 1 and phase 3.
template <int NCOL>
__device__ __forceinline__ void advance_interval(float ta, float dr, float E,
                                                 float* y, float* p) {
  int ns = (ta < 1.0f) ? 4 : ((ta < 8.0f) ? 2 : 1);
  float h = dr / (float)ns;
  for (int s = 0; s < ns; ++s) {
    float t  = ta + (float)s * h;
    float c1 = -2.0f / t - E;
    float tm = t + 0.5f * h;
    float c2 = -2.0f / tm - E;
    float te = t + h;
    float c3 = -2.0f / te - E;
#pragma unroll
    for (int c = 0; c < NCOL; ++c) rk4_col(c1, c2, c3, h, y[c], p[c]);
  }
}

// ---------------------------------------------------------------------------
// Stage the (at most two) rows of r_dsc this block needs into LDS with the
// CDNA5 async global->LDS path. Lanes 0..127 fetch the row for kfirst,
// lanes 128..255 the row for klast (when different). One burst, one
// s_wait_asynccnt, then a block barrier before anyone reads sm_r.
// ---------------------------------------------------------------------------
__device__ __forceinline__ void stage_r_rows(const float* __restrict__ r_dsc,
                                             float (&sm_r)[2][RROW],
                                             int kfirst, int klast, int tid) {
  int row  = tid >> 7;                  // 0: lanes 0..127, 1: lanes 128..255
  int lane = tid & 127;
  int k    = (row == 0) ? kfirst : klast;
  int idx  = k * CHUNK + lane;
  if (lane <= CHUNK && idx <= RN - 1 && (row == 0 || klast != kfirst)) {
    uint32_t lds = (uint32_t)(uintptr_t)&sm_r[row][lane];
    int off = idx * 4;                  // byte offset into r_dsc
    asm volatile("global_load_async_to_lds_b32 %0, %1, %2"
                 :: "v"(lds), "v"(off), "s"(r_dsc) : "memory");
  }
}

// ---------------------------------------------------------------------------
// Phase 1: per (chunk k, eigenvalue e) compute the 2x2 fundamental matrix
// Phi_k(E) mapping the state at r[k*CHUNK] to the state at chunk end.
// 100k independent threads -> ~3128 wave32 waves. The serial inner loop
// reads r from LDS (staged asynchronously), not from global memory.
// ---------------------------------------------------------------------------
__global__ void __launch_bounds__(256)
phase1_propagators(const float* __restrict__ evl,
                   const float* __restrict__ r_dsc,
                   float* __restrict__ phi) {
  __shared__ __align__(16) float sm_r[2][RROW];

  int tid = threadIdx.x;
  int id0 = blockIdx.x * 256;
  int id  = id0 + tid;

  int kfirst = id0 / NE;
  int klast  = min(id0 + 255, NPAIR - 1) / NE;

  stage_r_rows(r_dsc, sm_r, kfirst, klast, tid);
  asm volatile("s_wait_asynccnt 0x0" ::: "memory");
  __syncthreads();

  if (id >= NPAIR) return;
  int k = id / NE;
  int e = id - k * NE;
  int row = (k == kfirst) ? 0 : 1;
  float E = evl[e];

  int base  = k * CHUNK;
  int steps = min(CHUNK, (RN - 1) - base);

  // Columns of the fundamental matrix: (1,0) and (0,1) initial states.
  float y[2] = {1.0f, 0.0f};
  float p[2] = {0.0f, 1.0f};

  float ra = sm_r[row][0];              // wave-uniform -> LDS broadcast reads
  for (int j = 0; j < steps; ++j) {
    float rb = sm_r[row][j + 1];
    advance_interval<2>(ra, rb - ra, E, y, p);
    ra = rb;
  }
  float4 m = make_float4(y[0], p[0], y[1], p[1]);   // (u, u', v, v')
  *(float4*)&phi[(size_t)id * 4] = m;
}

// ---------------------------------------------------------------------------
// Phase 2: per eigenvalue, sequentially compose the NCHUNK propagators
// (the only remaining serial chain: length 100, not 10000). Also emits
// output row 0 = y0. Prefetches the next Phi row (global_prefetch_b8).
// ---------------------------------------------------------------------------
__global__ void __launch_bounds__(256)
phase2_scan(const float* __restrict__ phi,
            float* __restrict__ sstate,
            float* __restrict__ out) {
  int e = blockIdx.x * blockDim.x + threadIdx.x;
  if (e >= NE) return;

  // odeint returns y(t[0]) = y0 as the first output row.
  *(float2*)&out[(size_t)e * 2] = make_float2(0.0f, 1.0f);

  float s0 = 0.0f, s1 = 1.0f;           // y0 = (0, 1)
  for (int k = 0; k < NCHUNK; ++k) {
    *(float2*)&sstate[(size_t)(k * NE + e) * 2] = make_float2(s0, s1);
    if (k + 1 < NCHUNK)
      __builtin_prefetch(&phi[(size_t)((k + 1) * NE + e) * 4], 0, 0);
    float4 m = *(const float4*)&phi[(size_t)(k * NE + e) * 4];
    float n0 = m.x * s0 + m.z * s1;     // [u v; u' v'] * [s0; s1]
    float n1 = m.y * s0 + m.w * s1;
    s0 = n0;
    s1 = n1;
  }
}

// ---------------------------------------------------------------------------
// Phase 3: per (chunk, eigenvalue) re-integrate within the chunk from its
// start state, emitting the state at every interval end. Stores are float2
// with consecutive eigenvalues adjacent -> fully coalesced 2KB rows.
// r rows, eigenvalues and chunk-start states are all staged through LDS
// with the async global->LDS path (one burst, one ASYNCcnt wait).
// ---------------------------------------------------------------------------
__global__ void __launch_bounds__(256)
phase3_emit(const float* __restrict__ evl,
            const float* __restrict__ r_dsc,
            const float* __restrict__ sstate,
            float* __restrict__ out) {
  __shared__ __align__(16) float sm_r[2][RROW];
  __shared__ float               sm_evl[256];
  __shared__ __align__(8) float  sm_st[512];

  int tid = threadIdx.x;
  int id0 = blockIdx.x * 256;
  int id  = id0 + tid;
  bool active = (id < NPAIR);

  int kfirst = id0 / NE;
  int klast  = min(id0 + 255, NPAIR - 1) / NE;

  stage_r_rows(r_dsc, sm_r, kfirst, klast, tid);

  int k = 0, e = 0;
  if (active) {
    k = id / NE;
    e = id - k * NE;
    uint32_t lds_e = (uint32_t)(uintptr_t)&sm_evl[tid];
    uint32_t lds_s = (uint32_t)(uintptr_t)&sm_st[2 * tid];
    int off_e = e * 4;                         // byte offset into evl
    int off_s = (k * NE + e) * 8;              // byte offset into sstate
    asm volatile("global_load_async_to_lds_b32 %0, %1, %2"
                 :: "v"(lds_e), "v"(off_e), "s"(evl) : "memory");
    asm volatile("global_load_async_to_lds_b64 %0, %1, %2"
                 :: "v"(lds_s), "v"(off_s), "s"(sstate) : "memory");
  }
  asm volatile("s_wait_asynccnt 0x0" ::: "memory");
  __syncthreads();
  if (!active) return;

  int row = (k == kfirst) ? 0 : 1;
  float E    = sm_evl[tid];
  float y[1] = {sm_st[2 * tid]};
  float p[1] = {sm_st[2 * tid + 1]};

  int base  = k * CHUNK;
  int steps = min(CHUNK, (RN - 1) - base);

  float ra = sm_r[row][0];
  for (int j = 0; j < steps; ++j) {
    float rb = sm_r[row][j + 1];
    advance_interval<1>(ra, rb - ra, E, y, p);
    ra = rb;
    int gi = base + j + 1;
    *(float2*)&out[(size_t)gi * (NE * 2) + (size_t)e * 2] =
        make_float2(y[0], p[0]);
  }
}

// ---------------------------------------------------------------------------
extern "C" void kernel_launch(void* const* d_in, const int* in_sizes, int n_in,
                              void* d_out, int out_size, void* d_ws, size_t ws_size,
                              hipStream_t stream) {
  const float* evl   = (const float*)d_in[0];   // (1000,) f32
  const float* r_dsc = (const float*)d_in[1];   // (10000,) f32
  float* out = (float*)d_out;                   // (10000, 1000, 2) f32

  // Workspace layout (2.4 MB, fully rewritten each call):
  //   phi    : NCHUNK*NE*4 f32  (chunk propagators)
  //   sstate : NCHUNK*NE*2 f32  (chunk-start states)
  float* phi = (float*)d_ws;
  float* sst = phi + (size_t)NPAIR * 4;

  const int B = 256;                            // 8 wave32 waves per block
  phase1_propagators<<<(NPAIR + B - 1) / B, B, 0, stream>>>(evl, r_dsc, phi);
  phase2_scan<<<(NE + B - 1) / B, B, 0, stream>>>(phi, sst, out);
  phase3_emit<<<(NPAIR + B - 1) / B, B, 0, stream>>>(evl, r_dsc, sst, out);
}